// FRGCNModel_30932354466236
// MI455X (gfx1250) — compile-verified
//
#include <hip/hip_runtime.h>

#define NN 100000      // nodes
#define NE 3200000     // edges
#define NR 4           // relations
#define NG 512         // graphs
#define H  80          // hidden
#define NTILES ((NN + 15) / 16)   // 6250 exactly

typedef float    v2f __attribute__((ext_vector_type(2)));
typedef float    v8f __attribute__((ext_vector_type(8)));
typedef unsigned v4u __attribute__((ext_vector_type(4)));
typedef unsigned v8u __attribute__((ext_vector_type(8)));

// ---------------------------------------------------------------------------
// Fused dense kernel: per block of 4 node-tiles (64 nodes):
//   xw[n, m, :]  = X[n,:] @ Wrel[m]        (m = 0..NR-1)
//   hinit[n, :]  = X[n,:] @ root + bias    (m = NR)
// A fragment cached in VGPRs (X read once). B matrix staged in LDS by the
// Tensor Data Mover: wave 0 issues tensor_load_to_lds with a 2-D D#
// (tile = whole [Kreal x 80] f32 matrix), waits TENSORcnt, barrier publishes.
// LDS layout is TDM-natural row-major [k][h]; K-pad rows zeroed once.
// WMMA layouts:
//   A 16x4 f32: lanes 0-15 -> M=lane, regs = K 0/1 ; lanes 16-31 -> K 2/3.
//   C/D:        VGPR i: lanes 0-15 -> M=i, lanes 16-31 -> M=8+i, N=lane&15.
// ---------------------------------------------------------------------------
template <int KDIM>
__global__ __launch_bounds__(128) void k_dense(
    const float* __restrict__ X,     // [NN, KDIM]  (zero-padded in K)
    const float* __restrict__ Wrel,  // [NR, Kreal, H]
    const float* __restrict__ root,  // [Kreal, H]
    const float* __restrict__ bias,  // [H]
    int Kreal,
    float* __restrict__ xw,          // [NN, NR, H]
    float* __restrict__ hinit)       // [NN, H]
{
  __shared__ float Bs[KDIM * H];     // [k][h], K=80 -> 25.6 KB

  const int t    = threadIdx.x;
  const int tile = blockIdx.x * 4 + (t >> 5);
  const bool active = (tile < NTILES);
  const int lane = t & 31;
  const int half = lane >> 4;
  const int l    = lane & 15;

  // Zero the K-padding rows (k in [Kreal, KDIM)) once; TDM never writes them.
  for (int idx = Kreal * H + t; idx < KDIM * H; idx += 128) Bs[idx] = 0.0f;

  // Cache A fragment in registers (read X once per block).
  v2f av[KDIM / 4];
  if (active) {
    const float* __restrict__ xrow = X + (size_t)(tile * 16 + l) * KDIM;
#pragma unroll
    for (int s = 0; s < KDIM / 4; ++s)
      av[s] = *(const v2f*)(xrow + 4 * s + 2 * half);
  }

  const unsigned ldsAddr = (unsigned)(uintptr_t)&Bs[0];

  for (int m = 0; m <= NR; ++m) {
    const float* __restrict__ Wm =
        (m < NR) ? (Wrel + (size_t)m * Kreal * H) : root;
    __syncthreads();                 // previous m's reads done before DMA
    if (t < 32) {                    // one wave issues the TDM descriptor
      const unsigned long long ga = (unsigned long long)(uintptr_t)Wm;
      // D# group 0: count=1 (valid user desc), lds_addr, global_addr, type=2
      const v4u g0 = { 1u,
                       ldsAddr,
                       (unsigned)(ga & 0xFFFFFFFFu),
                       (unsigned)((ga >> 32) & 0x01FFFFFFu) | (2u << 30) };
      // D# group 1: data_size=4B; tensor 80 x Kreal; tile 80 x Kreal;
      // tensor_dim0_stride = 80. (2-D: groups 2/3 = NULL)
      const v8u g1 = { 0x00020000u,
                       (unsigned)H << 16,       // tensor_dim0[15:0] @ bits 63:48
                       (unsigned)Kreal << 16,   // tensor_dim1[15:0] @ bits 95:80
                       (unsigned)H << 16,       // tile_dim0        @ bits 127:112
                       (unsigned)Kreal,         // tile_dim1        @ bits 143:128
                       (unsigned)H,             // tensor_dim0_stride (48b) low
                       0u, 0u };
      asm volatile("tensor_load_to_lds %0, %1" :: "s"(g0), "s"(g1) : "memory");
      __builtin_amdgcn_s_wait_tensorcnt(0);
    }
    __syncthreads();                 // Bs visible to all waves

    if (active) {
#pragma unroll
      for (int ht = 0; ht < 5; ++ht) {
        const int col = ht * 16 + l;
        v8f acc = {};
#pragma unroll
        for (int s = 0; s < KDIM / 4; ++s) {
          const int ka = 4 * s + 2 * half;
          v2f b;
          b.x = Bs[ka * H + col];
          b.y = Bs[(ka + 1) * H + col];
          acc = __builtin_amdgcn_wmma_f32_16x16x4_f32(
              false, av[s], false, b, (short)0, acc, false, false);
        }
        if (m < NR) {
#pragma unroll
          for (int i = 0; i < 8; ++i)
            xw[((size_t)(tile * 16 + i + 8 * half) * NR + m) * H + col] = acc[i];
        } else {
          const float bv = bias[col];
#pragma unroll
          for (int i = 0; i < 8; ++i)
            hinit[(size_t)(tile * 16 + i + 8 * half) * H + col] = acc[i] + bv;
        }
      }
    }
  }
}

// out[512,15] = pooled[512,80] @ lin_w[80,15] + lin_b  (wT padded to [16,80])
__global__ __launch_bounds__(128) void k_lin(const float* __restrict__ pooled,
                                             const float* __restrict__ wT, // [16,80]
                                             const float* __restrict__ b,
                                             float* __restrict__ out)
{
  const int tile = blockIdx.x * 4 + (threadIdx.x >> 5);  // 32 tiles of 16 graphs
  if (tile * 16 >= NG) return;
  const int lane = threadIdx.x & 31, half = lane >> 4, l = lane & 15;
  const float* __restrict__ prow = pooled + (size_t)(tile * 16 + l) * H;
  const float* __restrict__ wrow = wT + (size_t)l * H;
  v8f acc = {};
#pragma unroll
  for (int s = 0; s < H / 4; ++s) {
    const int ka = 4 * s + 2 * half;
    const v2f a  = *(const v2f*)(prow + ka);
    const v2f bb = *(const v2f*)(wrow + ka);
    acc = __builtin_amdgcn_wmma_f32_16x16x4_f32(false, a, false, bb, (short)0,
                                                acc, false, false);
  }
  if (l < 15) {
    const float bv = b[l];
#pragma unroll
    for (int i = 0; i < 8; ++i)
      out[(size_t)(tile * 16 + i + 8 * half) * 15 + l] = acc[i] + bv;
  }
}

// xpad[n, 0:6] = x[n, :], xpad[n, 6:8] = 0
__global__ void k_padx(const float* __restrict__ x, float* __restrict__ xpad)
{
  const int i = blockIdx.x * blockDim.x + threadIdx.x;  // NN*8
  if (i >= NN * 8) return;
  const int n = i >> 3, k = i & 7;
  xpad[i] = (k < 6) ? x[n * 6 + k] : 0.0f;
}

// wT[j, k] = lin_w[k, j] for j<15 else 0   (j in [0,16), k in [0,80))
__global__ void k_padlin(const float* __restrict__ w, float* __restrict__ wT)
{
  const int i = blockIdx.x * blockDim.x + threadIdx.x;  // 16*80
  if (i >= 16 * H) return;
  const int j = i / H, k = i % H;
  wT[i] = (j < 15) ? w[k * 15 + j] : 0.0f;
}

__global__ void k_counts(const int* __restrict__ dst, const int* __restrict__ etype,
                         float* __restrict__ counts)
{
  const int e = blockIdx.x * blockDim.x + threadIdx.x;
  if (e < NE) atomicAdd(&counts[(size_t)dst[e] * NR + etype[e]], 1.0f);
}

// per edge: hout[dst,:] += xw[src, rel, :] / max(counts[dst,rel],1)
// 20 threads per edge, float4 each
__global__ void k_edges(const int* __restrict__ src, const int* __restrict__ dst,
                        const int* __restrict__ etype,
                        const float* __restrict__ counts,
                        const float* __restrict__ xw,
                        float* __restrict__ hout)
{
  const long long t = (long long)blockIdx.x * blockDim.x + threadIdx.x;
  const long long e = t / 20;
  const int       q = (int)(t % 20);
  if (e >= NE) return;
  const int s = src[e], d = dst[e], r = etype[e];
  const float norm = 1.0f / fmaxf(counts[(size_t)d * NR + r], 1.0f);
  const float4 m = *(const float4*)(xw + ((size_t)s * NR + r) * H + q * 4);
  float* o = hout + (size_t)d * H + q * 4;
  atomicAdd(o + 0, m.x * norm);
  atomicAdd(o + 1, m.y * norm);
  atomicAdd(o + 2, m.z * norm);
  atomicAdd(o + 3, m.w * norm);
}

__global__ void k_relu(float* __restrict__ h, long n)
{
  const long i = (long)blockIdx.x * blockDim.x + threadIdx.x;
  if (i < n) h[i] = fmaxf(h[i], 0.0f);
}

__global__ void k_pool(const float* __restrict__ h, const int* __restrict__ batch,
                       float* __restrict__ pooled)
{
  const long long t = (long long)blockIdx.x * blockDim.x + threadIdx.x;
  const long long n = t / 20;
  const int       q = (int)(t % 20);
  if (n >= NN) return;
  const int g = batch[n];
  const float4 v = *(const float4*)(h + n * H + q * 4);
  float* o = pooled + (size_t)g * H + q * 4;
  atomicAdd(o + 0, v.x);
  atomicAdd(o + 1, v.y);
  atomicAdd(o + 2, v.z);
  atomicAdd(o + 3, v.w);
}

extern "C" void kernel_launch(void* const* d_in, const int* in_sizes, int n_in,
                              void* d_out, int out_size, void* d_ws, size_t ws_size,
                              hipStream_t stream) {
  (void)in_sizes; (void)n_in; (void)out_size; (void)ws_size;
  const float* x     = (const float*)d_in[0];
  const int*   ei    = (const int*)d_in[1];   // [2, NE]
  const int*   etype = (const int*)d_in[2];
  const int*   batch = (const int*)d_in[3];
  const float* W1    = (const float*)d_in[4];
  const float* root1 = (const float*)d_in[5];
  const float* b1    = (const float*)d_in[6];
  const float* W2    = (const float*)d_in[7];
  const float* root2 = (const float*)d_in[8];
  const float* b2    = (const float*)d_in[9];
  const float* lin_w = (const float*)d_in[10];
  const float* lin_b = (const float*)d_in[11];
  float*       out   = (float*)d_out;

  const int* src = ei;
  const int* dst = ei + NE;

  // Workspace carve-out (~197 MB total)
  char* ws = (char*)d_ws;
  size_t off = 0;
  auto carve = [&](size_t bytes) {
    void* p = ws + off;
    off = (off + bytes + 255) & ~(size_t)255;
    return p;
  };
  float* counts = (float*)carve((size_t)NN * NR * 4);
  float* xw     = (float*)carve((size_t)NN * NR * H * 4);   // reused both layers
  float* h1     = (float*)carve((size_t)NN * H * 4);
  float* h2     = (float*)carve((size_t)NN * H * 4);
  float* pooled = (float*)carve((size_t)NG * H * 4);
  float* xpad   = (float*)carve((size_t)NN * 8 * 4);
  float* wlinT  = (float*)carve((size_t)16 * H * 4);

  hipMemsetAsync(counts, 0, (size_t)NN * NR * 4, stream);
  hipMemsetAsync(pooled, 0, (size_t)NG * H * 4, stream);

  k_padx  <<<(NN * 8 + 255) / 256, 256, 0, stream>>>(x, xpad);
  k_padlin<<<(16 * H + 255) / 256, 256, 0, stream>>>(lin_w, wlinT);
  k_counts<<<(NE + 255) / 256, 256, 0, stream>>>(dst, etype, counts);

  const int dBlocks = (NTILES + 3) / 4;                       // 1563
  const long long eThreads = (long long)NE * 20;
  const int eBlocks = (int)((eThreads + 255) / 256);
  const long hn = (long)NN * H;
  const int rBlocks = (int)((hn + 255) / 256);
  const long long pThreads = (long long)NN * 20;
  const int pBlocks = (int)((pThreads + 255) / 256);

  // ---- Layer 1 (K padded 6 -> 8) ----
  k_dense<8><<<dBlocks, 128, 0, stream>>>(xpad, W1, root1, b1, 6, xw, h1);
  k_edges<<<eBlocks, 256, 0, stream>>>(src, dst, etype, counts, xw, h1);
  k_relu<<<rBlocks, 256, 0, stream>>>(h1, hn);

  // ---- Layer 2 (K = 80) ----
  k_dense<80><<<dBlocks, 128, 0, stream>>>(h1, W2, root2, b2, 80, xw, h2);
  k_edges<<<eBlocks, 256, 0, stream>>>(src, dst, etype, counts, xw, h2);
  k_relu<<<rBlocks, 256, 0, stream>>>(h2, hn);

  // ---- Pool + head ----
  k_pool<<<pBlocks, 256, 0, stream>>>(h2, batch, pooled);
  k_lin<<<8, 128, 0, stream>>>(pooled, wlinT, lin_b, out);
}